// LiSMCA_Bridge_46265387712745
// MI455X (gfx1250) — compile-verified
//
#include <hip/hip_runtime.h>
#include <hip/hip_bf16.h>
#include <math.h>

#define B_ 8
#define DH_ 1024
#define P_ 64
#define V_ 50257
#define NP_ 256
#define KTOP_ 8
#define SKEW_ 2016

typedef float v2f __attribute__((ext_vector_type(2)));
typedef float v8f __attribute__((ext_vector_type(8)));

static __device__ __forceinline__ v8f wmma_f32_4(v2f a, v2f b, v8f c) {
  // D = A(16x4,f32) * B(4x16,f32) + C(16x16,f32)  -> v_wmma_f32_16x16x4_f32
  return __builtin_amdgcn_wmma_f32_16x16x4_f32(false, a, false, b, (short)0, c, false, false);
}

// ---------------------------------------------------------------------------
// K1: Cayley transform  Q = (I - A)^-1 (I + A), Gauss-Jordan in LDS, 1 block/batch
// ---------------------------------------------------------------------------
__global__ void k_cayley(const float* __restrict__ A, float* __restrict__ Q) {
  __shared__ float Ms[64][65];
  __shared__ float Xs[64][65];
  int b = blockIdx.x;
  int i = threadIdx.x;  // 64 threads, one row each
  for (int j = 0; j < 64; ++j) {
    float a = A[(size_t)b * 4096 + i * 64 + j];
    float id = (i == j) ? 1.f : 0.f;
    Ms[i][j] = id - a;
    Xs[i][j] = id + a;
  }
  __syncthreads();
  for (int k = 0; k < 64; ++k) {
    if (i == k) {
      float pinv = 1.f / Ms[k][k];
      for (int j = 0; j < 64; ++j) { Ms[k][j] *= pinv; Xs[k][j] *= pinv; }
    }
    __syncthreads();
    if (i != k) {
      float f = Ms[i][k];
      for (int j = 0; j < 64; ++j) { Ms[i][j] -= f * Ms[k][j]; Xs[i][j] -= f * Xs[k][j]; }
    }
    __syncthreads();
  }
  for (int j = 0; j < 64; ++j) Q[(size_t)b * 4096 + i * 64 + j] = Xs[i][j];
}

// ---------------------------------------------------------------------------
// K2: q_t = h@WQ + b, h_lie = h@hlie_w + b   (8 blocks x 64 threads)
// ---------------------------------------------------------------------------
__global__ void k_vecs(const float* __restrict__ h, const float* __restrict__ WQ,
                       const float* __restrict__ WQb, const float* __restrict__ HL,
                       const float* __restrict__ HLb, float* __restrict__ qv,
                       float* __restrict__ hl) {
  int b = blockIdx.x, p = threadIdx.x;
  float sq = WQb[p], sh = HLb[p];
  const float* hb = h + (size_t)b * DH_;
  for (int d = 0; d < DH_; ++d) {
    float hv = hb[d];
    sq += hv * WQ[d * P_ + p];
    sh += hv * HL[d * P_ + p];
  }
  qv[b * P_ + p] = sq;
  hl[b * P_ + p] = sh;
}

// ---------------------------------------------------------------------------
// K3: generic f32 WMMA GEMM  C[b] = A[b](MxK) * B[b](KxN) (+bias)
// one wave per 16x16 output tile; M,N multiples of 16, K multiple of 4
// ---------------------------------------------------------------------------
__global__ void k_wmma_gemm(const float* __restrict__ A, const float* __restrict__ Bm,
                            const float* __restrict__ bias, float* __restrict__ C,
                            int N, int Kd, long sA, long sB, long sC, int ntiles) {
  int lane = threadIdx.x;
  int tile = blockIdx.x;
  int b = blockIdx.y;
  int mt = tile / ntiles, nt = tile % ntiles;
  const float* Ab = A + (size_t)b * sA;
  const float* Bb = Bm + (size_t)b * sB;
  float* Cb = C + (size_t)b * sC;
  int row0 = mt * 16, col0 = nt * 16;
  int m = lane & 15, kb = (lane >> 4) * 2;
  v8f acc = {};
  for (int k = 0; k < Kd; k += 4) {
    v2f a, bb;
    a.x = Ab[(size_t)(row0 + m) * Kd + k + kb];
    a.y = Ab[(size_t)(row0 + m) * Kd + k + kb + 1];
    bb.x = Bb[(size_t)(k + kb) * N + col0 + m];
    bb.y = Bb[(size_t)(k + kb + 1) * N + col0 + m];
    acc = wmma_f32_4(a, bb, acc);
  }
  float bv = bias ? bias[col0 + m] : 0.f;
  int rofs = (lane >= 16) ? 8 : 0;
#pragma unroll
  for (int r = 0; r < 8; ++r) {
    Cb[(size_t)(row0 + r + rofs) * N + col0 + m] = acc[r] + bv;
  }
}

// ---------------------------------------------------------------------------
// K5: attention: scores = 0.125*q.K - 0.5*(rowdot(T,B) - 2*(Gh).B + hGh),
// softmax over n, c_t = alpha @ V_t.   1 block/batch, 256 threads (thread = n)
// ---------------------------------------------------------------------------
__global__ void k_attention(const float* __restrict__ qv, const float* __restrict__ hl,
                            const float* __restrict__ G, const float* __restrict__ Brot,
                            const float* __restrict__ Tbuf, const float* __restrict__ Kt,
                            const float* __restrict__ Vt, float* __restrict__ ct) {
  __shared__ float hls[64], qs[64], us[64], sc[256], red[256];
  __shared__ float chh_s, mx_s, Z_s;
  int b = blockIdx.x, tid = threadIdx.x;
  if (tid < 64) { hls[tid] = hl[b * 64 + tid]; qs[tid] = qv[b * 64 + tid]; }
  __syncthreads();
  if (tid < 64) {
    float u = 0.f;
    const float* Gr = G + (size_t)b * 4096 + tid * 64;
    for (int q = 0; q < 64; ++q) u += Gr[q] * hls[q];
    us[tid] = u;
  }
  __syncthreads();
  if (tid == 0) { float s = 0.f; for (int p = 0; p < 64; ++p) s += us[p] * hls[p]; chh_s = s; }
  __syncthreads();
  const float* Bn = Brot + (size_t)b * NP_ * 64 + tid * 64;
  const float* Tn = Tbuf + (size_t)b * NP_ * 64 + tid * 64;
  const float* Kn = Kt + (size_t)b * NP_ * 64 + tid * 64;
  float s1 = 0.f, s2 = 0.f, s3 = 0.f;
  for (int p = 0; p < 64; ++p) { float bv = Bn[p]; s1 += Tn[p] * bv; s2 += us[p] * bv; s3 += qs[p] * Kn[p]; }
  float score = 0.125f * s3 - 0.5f * (s1 - 2.f * s2 + chh_s);
  red[tid] = score;
  __syncthreads();
  for (int off = 128; off > 0; off >>= 1) { if (tid < off) red[tid] = fmaxf(red[tid], red[tid + off]); __syncthreads(); }
  if (tid == 0) mx_s = red[0];
  __syncthreads();
  float e = __expf(score - mx_s);
  sc[tid] = e;
  red[tid] = e;
  __syncthreads();
  for (int off = 128; off > 0; off >>= 1) { if (tid < off) red[tid] += red[tid + off]; __syncthreads(); }
  if (tid == 0) Z_s = red[0];
  __syncthreads();
  if (tid < 64) {
    float c = 0.f;
    const float* Vb = Vt + (size_t)b * NP_ * 64;
    for (int n = 0; n < NP_; ++n) c += sc[n] * Vb[n * 64 + tid];
    ct[b * 64 + tid] = c / Z_s;
  }
}

// ---------------------------------------------------------------------------
// K6: g_t (triu of A), g_proj = g_t@gdown, gq = sum S*gproj^2,
// c_rot = (c_t@WC + b)@Q, gamma = sigmoid([h,c,g]@wgamma + b). 1 block/batch.
// ---------------------------------------------------------------------------
__global__ void k_small(const float* __restrict__ h, const float* __restrict__ A,
                        const float* __restrict__ ct, const float* __restrict__ gdown,
                        const float* __restrict__ Sinv, const float* __restrict__ WC,
                        const float* __restrict__ WCb, const float* __restrict__ Qm,
                        const float* __restrict__ wg, const float* __restrict__ wgb,
                        float* __restrict__ gproj, float* __restrict__ gq,
                        float* __restrict__ crot, float* __restrict__ gamma) {
  __shared__ float gts[SKEW_];
  __shared__ float cts[64], ctmp[64], red[64];
  int b = blockIdx.x, t = threadIdx.x;  // 64 threads
  if (t < 63) {
    int base = t * 63 - t * (t - 1) / 2;
    const float* Ar = A + (size_t)b * 4096 + t * 64;
    for (int j = t + 1; j < 64; ++j) gts[base + (j - t - 1)] = Ar[j];
  }
  cts[t] = ct[b * 64 + t];
  __syncthreads();
  float gp = 0.f;
  for (int k = 0; k < SKEW_; ++k) gp += gts[k] * gdown[(size_t)k * 64 + t];
  gproj[b * 64 + t] = gp;
  red[t] = Sinv[t] * gp * gp;
  __syncthreads();
  if (t == 0) { float s = 0.f; for (int p = 0; p < 64; ++p) s += red[p]; gq[b] = s; }
  float cm = WCb[t];
  for (int q = 0; q < 64; ++q) cm += cts[q] * WC[q * 64 + t];
  ctmp[t] = cm;
  __syncthreads();
  float cr = 0.f;
  const float* Qb = Qm + (size_t)b * 4096;
  for (int q = 0; q < 64; ++q) cr += ctmp[q] * Qb[q * 64 + t];
  crot[b * 64 + t] = cr;
  float part = 0.f;
  for (int ii = t; ii < DH_ + 64 + SKEW_; ii += 64) {
    float x = (ii < DH_) ? h[(size_t)b * DH_ + ii]
              : (ii < DH_ + 64) ? cts[ii - DH_] : gts[ii - DH_ - 64];
    part += x * wg[ii];
  }
  __syncthreads();
  red[t] = part;
  __syncthreads();
  if (t == 0) {
    float s = wgb[0];
    for (int p = 0; p < 64; ++p) s += red[p];
    gamma[b] = 1.f / (1.f + __expf(-s));
  }
}

// ---------------------------------------------------------------------------
// K7: fused logits. One wave per 64-vocab tile:
//   - h (8x1024) staged into LDS via global_load_async_to_lds_b128 (ASYNCcnt)
//   - base = h@WO via WMMA, K=1024; one A frag feeds 4 vocab subtiles/k-step
//   - WO stream prefetched 8 rows ahead (global_prefetch_b8)
//   - per 16-v subtile: eg = E@WG via WMMA, kappa quadratic, combine -> d_out
// ---------------------------------------------------------------------------
__global__ void k_fused_logits(const float* __restrict__ h, const float* __restrict__ WO,
                               const float* __restrict__ E, const float* __restrict__ WG,
                               const float* __restrict__ WGb, const float* __restrict__ Sinv,
                               const float* __restrict__ gproj, const float* __restrict__ crot,
                               const float* __restrict__ gamma, const float* __restrict__ gq,
                               float* __restrict__ dout) {
  __shared__ float hs[B_ * DH_];  // 32 KB, filled by async DMA
  __shared__ float egs[16 * 64];
  __shared__ float gprs[512], crs[512], Ss[64], WGbs[64];
  __shared__ float gms[8], gqs[8];
  __shared__ float sv2[32];
  __shared__ float kd2[2][16][8];
  __shared__ float ce2[2][16][8];
  int lane = threadIdx.x;  // 32 = one wave
  int v0 = blockIdx.x * 64;

  // ---- stage h into LDS with async global->LDS DMA (b128 per lane)
  {
    unsigned lbase = (unsigned)(uintptr_t)(&hs[0]);
    for (int it = 0; it < (B_ * DH_) / 128; ++it) {
      int elt = it * 128 + lane * 4;
      unsigned laddr = lbase + (unsigned)(elt * 4);
      unsigned long long gaddr = (unsigned long long)(const void*)(h + elt);
      asm volatile("global_load_async_to_lds_b128 %0, %1, off"
                   :: "v"(laddr), "v"(gaddr)
                   : "memory");
    }
  }
  for (int i = lane; i < 512; i += 32) { gprs[i] = gproj[i]; crs[i] = crot[i]; }
  for (int i = lane; i < 64; i += 32) { Ss[i] = Sinv[i]; WGbs[i] = WGb[i]; }
  if (lane < 8) { gms[lane] = gamma[lane]; gqs[lane] = gq[lane]; }
  asm volatile("s_wait_asynccnt 0x0" ::: "memory");
  __syncthreads();

  int m = lane & 15, kb2 = (lane >> 4) * 2;
  int mr = m & 7;  // clamped row for LDS read; selected to 0 when m >= 8
  // ---- base logits: 4 vocab subtiles share one A fragment per k-step
  v8f accB0 = {}, accB1 = {}, accB2 = {}, accB3 = {};
  for (int k = 0; k < DH_; k += 4) {
    v2f a;
    float h0 = hs[mr * DH_ + k + kb2];
    float h1 = hs[mr * DH_ + k + kb2 + 1];
    a.x = (m < 8) ? h0 : 0.f;
    a.y = (m < 8) ? h1 : 0.f;
    int pr = (k + 8 < DH_) ? (k + 8) : (DH_ - 4);
    __builtin_prefetch(WO + (size_t)pr * V_ + v0 + 16 * (lane & 3), 0, 0);
#pragma unroll
    for (int t = 0; t < 4; ++t) {
      int vt = v0 + t * 16 + m;
      bool vok = vt < V_;
      v2f bb;
      bb.x = vok ? WO[(size_t)(k + kb2) * V_ + vt] : 0.f;
      bb.y = vok ? WO[(size_t)(k + kb2 + 1) * V_ + vt] : 0.f;
      v8f acc = (t == 0) ? accB0 : (t == 1) ? accB1 : (t == 2) ? accB2 : accB3;
      acc = wmma_f32_4(a, bb, acc);
      if (t == 0) accB0 = acc; else if (t == 1) accB1 = acc;
      else if (t == 2) accB2 = acc; else accB3 = acc;
    }
  }

  // ---- per 16-v subtile: E_gen tile via WMMA + kappa/group-logit combine
#pragma unroll
  for (int t = 0; t < 4; ++t) {
    int vt = v0 + t * 16 + m;
    bool vok = vt < V_;
    v8f e0 = {}, e1 = {}, e2 = {}, e3 = {};
    for (int k = 0; k < 64; k += 4) {
      v2f a;
      a.x = vok ? E[(size_t)vt * 64 + k + kb2] : 0.f;
      a.y = vok ? E[(size_t)vt * 64 + k + kb2 + 1] : 0.f;
      v2f b0, b1, b2, b3;
      b0.x = WG[(k + kb2) * 64 + m];       b0.y = WG[(k + kb2 + 1) * 64 + m];
      b1.x = WG[(k + kb2) * 64 + 16 + m];  b1.y = WG[(k + kb2 + 1) * 64 + 16 + m];
      b2.x = WG[(k + kb2) * 64 + 32 + m];  b2.y = WG[(k + kb2 + 1) * 64 + 32 + m];
      b3.x = WG[(k + kb2) * 64 + 48 + m];  b3.y = WG[(k + kb2 + 1) * 64 + 48 + m];
      e0 = wmma_f32_4(a, b0, e0);
      e1 = wmma_f32_4(a, b1, e1);
      e2 = wmma_f32_4(a, b2, e2);
      e3 = wmma_f32_4(a, b3, e3);
    }
    int rofs = (lane >= 16) ? 8 : 0;
#pragma unroll
    for (int r = 0; r < 8; ++r) {
      int row = r + rofs;
      egs[row * 64 + m]      = e0[r] + WGbs[m];
      egs[row * 64 + 16 + m] = e1[r] + WGbs[16 + m];
      egs[row * 64 + 32 + m] = e2[r] + WGbs[32 + m];
      egs[row * 64 + 48 + m] = e3[r] + WGbs[48 + m];
    }
    __syncthreads();
    int half = lane >> 4;
    float sv = 0.f;
    float kd[8] = {0, 0, 0, 0, 0, 0, 0, 0};
    float ce[8] = {0, 0, 0, 0, 0, 0, 0, 0};
    for (int pp = 0; pp < 32; ++pp) {
      int p = half * 32 + pp;
      float eg = egs[m * 64 + p];
      float se = Ss[p] * eg;
      sv += se * eg;
      float Ev = vok ? E[(size_t)vt * 64 + p] : 0.f;
#pragma unroll
      for (int q = 0; q < 8; ++q) {
        kd[q] += se * gprs[q * 64 + p];
        ce[q] += crs[q * 64 + p] * Ev;
      }
    }
    sv2[half * 16 + m] = sv;
#pragma unroll
    for (int q = 0; q < 8; ++q) { kd2[half][m][q] = kd[q]; ce2[half][m][q] = ce[q]; }
    __syncthreads();
    if (lane < 16 && vok) {
      float svt = sv2[m] + sv2[16 + m];
      v8f accT = (t == 0) ? accB0 : (t == 1) ? accB1 : (t == 2) ? accB2 : accB3;
#pragma unroll
      for (int q = 0; q < 8; ++q) {
        float kdot = kd2[0][m][q] + kd2[1][m][q];
        float cE = ce2[0][m][q] + ce2[1][m][q];
        float kappa = -svt + 2.f * kdot - gqs[q];
        float gl = 0.125f * cE + kappa;
        dout[(size_t)q * V_ + vt] = accT[q] + gms[q] * gl;
      }
    }
    __syncthreads();
  }
}

// ---------------------------------------------------------------------------
// K8: softmax over V, entropy, top-8, E_exp, rho, mu, m_next, A_next.
// 1 block/batch, 256 threads. Reads logits from d_out, writes m_next/A_next.
// ---------------------------------------------------------------------------
__global__ void k_finalize(const float* __restrict__ h, const float* __restrict__ A,
                           const float* __restrict__ emask, const float* __restrict__ G,
                           const float* __restrict__ E, const float* __restrict__ Qm,
                           const float* __restrict__ hl, const float* __restrict__ ct,
                           const float* __restrict__ wmu, const float* __restrict__ wmub,
                           float* __restrict__ dout) {
  __shared__ float red1[256], red2[256];
  __shared__ float cand_v[256 * KTOP_];
  __shared__ int cand_i[256 * KTOP_];
  __shared__ float topp_s[KTOP_];
  __shared__ int topi_s[KTOP_];
  __shared__ float wqs[64], eexp[64], dvec[64], mns[64], ems[64], hls[64];
  __shared__ float mx_s, Z_s, H_s, rho_s, mu_s;
  int b = blockIdx.x, tid = threadIdx.x;
  float* lg = dout + (size_t)b * V_;
  // pass 1: max
  float m = -3.4e38f;
  for (int v = tid; v < V_; v += 256) m = fmaxf(m, lg[v]);
  red1[tid] = m;
  __syncthreads();
  for (int off = 128; off > 0; off >>= 1) { if (tid < off) red1[tid] = fmaxf(red1[tid], red1[tid + off]); __syncthreads(); }
  if (tid == 0) mx_s = red1[0];
  __syncthreads();
  float mx = mx_s;
  // pass 2: sum exp, sum exp*x, per-thread top-8
  float s1 = 0.f, s2 = 0.f;
  float tv[KTOP_]; int ti[KTOP_];
  for (int k = 0; k < KTOP_; ++k) { tv[k] = -3.4e38f; ti[k] = 0; }
  for (int v = tid; v < V_; v += 256) {
    float l = lg[v];
    float x = l - mx;
    float e = __expf(x);
    s1 += e; s2 += e * x;
    if (l > tv[KTOP_ - 1]) {
      int k = KTOP_ - 1;
      while (k > 0 && l > tv[k - 1]) { tv[k] = tv[k - 1]; ti[k] = ti[k - 1]; --k; }
      tv[k] = l; ti[k] = v;
    }
  }
  red1[tid] = s1; red2[tid] = s2;
  for (int k = 0; k < KTOP_; ++k) { cand_v[tid * KTOP_ + k] = tv[k]; cand_i[tid * KTOP_ + k] = ti[k]; }
  __syncthreads();
  for (int off = 128; off > 0; off >>= 1) {
    if (tid < off) { red1[tid] += red1[tid + off]; red2[tid] += red2[tid + off]; }
    __syncthreads();
  }
  if (tid == 0) { Z_s = red1[0]; H_s = logf(red1[0]) - red2[0] / red1[0]; }
  __syncthreads();
  float Z = Z_s;
  // top-8 global selection + topp
  if (tid == 0) {
    for (int kk = 0; kk < KTOP_; ++kk) {
      float best = -3.4e38f; int besti = 0; int bestslot = 0;
      for (int c = 0; c < 256 * KTOP_; ++c) {
        if (cand_v[c] > best) { best = cand_v[c]; besti = cand_i[c]; bestslot = c; }
      }
      topi_s[kk] = besti;
      topp_s[kk] = __expf(best - mx) / Z;
      cand_v[bestslot] = -3.4e38f;
    }
    float ps = 0.f;
    for (int k = 0; k < KTOP_; ++k) ps += topp_s[k];
    for (int k = 0; k < KTOP_; ++k) topp_s[k] /= ps;
  }
  __syncthreads();
  // E_exp = (sum_k topp_k * E[topi_k]) @ Q[b]
  if (tid < 64) {
    float w = 0.f;
    for (int k = 0; k < KTOP_; ++k) w += topp_s[k] * E[(size_t)topi_s[k] * 64 + tid];
    wqs[tid] = w;
    hls[tid] = hl[b * 64 + tid];
  }
  __syncthreads();
  if (tid < 64) {
    float ee = 0.f;
    const float* Qb = Qm + (size_t)b * 4096;
    for (int q = 0; q < 64; ++q) ee += wqs[q] * Qb[q * 64 + tid];
    eexp[tid] = ee;
    dvec[tid] = ee - hls[tid];
  }
  __syncthreads();
  if (tid < 64) {
    float tp = 0.f;
    const float* Gr = G + (size_t)b * 4096 + tid * 64;
    for (int q = 0; q < 64; ++q) tp += Gr[q] * dvec[q];
    red1[tid] = dvec[tid] * tp;
  }
  __syncthreads();
  if (tid == 0) { float s = 0.f; for (int p = 0; p < 64; ++p) s += red1[p]; rho_s = s; }
  __syncthreads();
  // mu = sigmoid([h, c_t, H, rho] @ wmu + b)
  float part = 0.f;
  for (int ii = tid; ii < DH_ + 64 + 2; ii += 256) {
    float x;
    if (ii < DH_) x = h[(size_t)b * DH_ + ii];
    else if (ii < DH_ + 64) x = ct[b * 64 + (ii - DH_)];
    else if (ii == DH_ + 64) x = H_s;
    else x = rho_s;
    part += x * wmu[ii];
  }
  red1[tid] = part;
  __syncthreads();
  for (int off = 128; off > 0; off >>= 1) { if (tid < off) red1[tid] += red1[tid + off]; __syncthreads(); }
  if (tid == 0) mu_s = 1.f / (1.f + __expf(-(red1[0] + wmub[0])));
  __syncthreads();
  float* mout = dout + (size_t)B_ * V_;
  float* aout = dout + (size_t)B_ * V_ + (size_t)B_ * 64;
  if (tid < 64) {
    float em = emask[b * 64 + tid];
    float mn = mu_s * em + (1.f - mu_s) * eexp[tid];
    mns[tid] = mn; ems[tid] = em;
    mout[b * 64 + tid] = mn;
  }
  __syncthreads();
  for (int idx = tid; idx < 4096; idx += 256) {
    int i = idx >> 6, j = idx & 63;
    aout[(size_t)b * 4096 + idx] =
        A[(size_t)b * 4096 + idx] + 0.1f * (mns[i] * ems[j] - mns[j] * ems[i]);
  }
}

// ---------------------------------------------------------------------------
extern "C" void kernel_launch(void* const* d_in, const int* in_sizes, int n_in,
                              void* d_out, int out_size, void* d_ws, size_t ws_size,
                              hipStream_t stream) {
  (void)in_sizes; (void)n_in; (void)out_size; (void)ws_size;
  const float* h     = (const float*)d_in[0];
  const float* A     = (const float*)d_in[1];
  const float* Bt    = (const float*)d_in[2];
  const float* E     = (const float*)d_in[3];
  const float* emask = (const float*)d_in[4];
  const float* G     = (const float*)d_in[5];
  // d_in[6] = z0 (unused by reference)
  const float* WQ    = (const float*)d_in[7];
  const float* WQb   = (const float*)d_in[8];
  const float* WK    = (const float*)d_in[9];
  const float* WKb   = (const float*)d_in[10];
  const float* WV    = (const float*)d_in[11];
  const float* WVb   = (const float*)d_in[12];
  const float* WC    = (const float*)d_in[13];
  const float* WCb   = (const float*)d_in[14];
  const float* WO    = (const float*)d_in[15];
  const float* WG    = (const float*)d_in[16];
  const float* WGb   = (const float*)d_in[17];
  const float* wg    = (const float*)d_in[18];
  const float* wgb   = (const float*)d_in[19];
  const float* wmu   = (const float*)d_in[20];
  const float* wmub  = (const float*)d_in[21];
  const float* gdown = (const float*)d_in[22];
  const float* Sinv  = (const float*)d_in[23];
  const float* HLw   = (const float*)d_in[24];
  const float* HLb   = (const float*)d_in[25];
  float* out = (float*)d_out;
  float* ws  = (float*)d_ws;

  float* Q     = ws;                 // 8*64*64    = 32768
  float* qv    = Q + 32768;          // 512
  float* hl    = qv + 512;           // 512
  float* Brot  = hl + 512;           // 8*256*64   = 131072
  float* Kt    = Brot + 131072;      // 131072
  float* Vt    = Kt + 131072;        // 131072
  float* Tb    = Vt + 131072;        // 131072
  float* ct    = Tb + 131072;        // 512
  float* gproj = ct + 512;           // 512
  float* crot  = gproj + 512;        // 512
  float* gamma = crot + 512;         // 8
  float* gq    = gamma + 8;          // 8   -> total ~2.24 MB

  k_cayley<<<dim3(B_), dim3(64), 0, stream>>>(A, Q);
  k_vecs<<<dim3(B_), dim3(64), 0, stream>>>(h, WQ, WQb, HLw, HLb, qv, hl);
  // B_rot = B_t @ Q   (batched 8, M=256 N=64 K=64; 16x4 tiles)
  k_wmma_gemm<<<dim3(16 * 4, B_), dim3(32), 0, stream>>>(Bt, Q, nullptr, Brot,
                                                         64, 64, 256L * 64, 64L * 64, 256L * 64, 4);
  // T = B_rot @ G_tilde (batched) for the omega quadratic form
  k_wmma_gemm<<<dim3(16 * 4, B_), dim3(32), 0, stream>>>(Brot, G, nullptr, Tb,
                                                         64, 64, 256L * 64, 64L * 64, 256L * 64, 4);
  // K_t / V_t = B_rot(2048x64 flat) @ W + b
  k_wmma_gemm<<<dim3(128 * 4, 1), dim3(32), 0, stream>>>(Brot, WK, WKb, Kt, 64, 64, 0, 0, 0, 4);
  k_wmma_gemm<<<dim3(128 * 4, 1), dim3(32), 0, stream>>>(Brot, WV, WVb, Vt, 64, 64, 0, 0, 0, 4);
  k_attention<<<dim3(B_), dim3(256), 0, stream>>>(qv, hl, G, Brot, Tb, Kt, Vt, ct);
  k_small<<<dim3(B_), dim3(64), 0, stream>>>(h, A, ct, gdown, Sinv, WC, WCb, Q, wg, wgb,
                                             gproj, gq, crot, gamma);
  k_fused_logits<<<dim3((V_ + 63) / 64), dim3(32), 0, stream>>>(h, WO, E, WG, WGb, Sinv,
                                                                gproj, crot, gamma, gq, out);
  k_finalize<<<dim3(B_), dim3(256), 0, stream>>>(h, A, emask, G, E, Q, hl, ct, wmu, wmub, out);
}